// DiscriminatorWithLS4_46368466927686
// MI455X (gfx1250) — compile-verified
//
#include <hip/hip_runtime.h>
#include <hip/hip_bf16.h>

// Problem dims (from reference)
#define C_IN   8
#define BATCH  64
#define TLEN   4096
#define DMODEL 128
#define NSTATE 64
#define HID    128
#define CB     (C_IN * BATCH)   // 512 combined (c,b) columns

typedef __attribute__((ext_vector_type(2))) float v2f;
typedef __attribute__((ext_vector_type(8))) float v8f;

// ---------------------------------------------------------------------------
// Kernel 1: per (d,n): G = B_ssm*C_ssm, L2A = log2(sigmoid(log_a))
// ---------------------------------------------------------------------------
__global__ void prep_kernel(const float* __restrict__ log_a,
                            const float* __restrict__ B_ssm,
                            const float* __restrict__ C_ssm,
                            float* __restrict__ G,
                            float* __restrict__ L2A) {
    int i = blockIdx.x * blockDim.x + threadIdx.x;  // over DMODEL*NSTATE = 8192
    if (i >= DMODEL * NSTATE) return;
    float la = log_a[i];
    float a  = 1.0f / (1.0f + __expf(-la));         // sigmoid, in (0,1)
    G[i]   = B_ssm[i] * C_ssm[i];
    L2A[i] = __log2f(a);                            // < 0
}

// ---------------------------------------------------------------------------
// Kernel 2: K[d,t] = sum_n G[d,n] * a[d,n]^(T-1-t)  via exp2 (no recurrence)
// One thread per (d,t); lanes share d -> coalesced writes, broadcast G/L2A.
// ---------------------------------------------------------------------------
__global__ void kfilter_kernel(const float* __restrict__ G,
                               const float* __restrict__ L2A,
                               float* __restrict__ Kmat) {
    int i = blockIdx.x * blockDim.x + threadIdx.x;  // i = d*TLEN + t
    int d = i >> 12;
    int t = i & (TLEN - 1);
    float e = (float)(TLEN - 1 - t);
    const float* g = G   + d * NSTATE;
    const float* l = L2A + d * NSTATE;
    float s = 0.0f;
#pragma unroll 8
    for (int n = 0; n < NSTATE; ++n) {
        s += g[n] * exp2f(e * l[n]);                // v_exp_f32 path
    }
    Kmat[i] = s;
}

// ---------------------------------------------------------------------------
// Kernel 3: Ksum[d] = sum_t K[d,t]  (only needed for b_in term; b_in==0 here
// but we keep it general). One block per d.
// ---------------------------------------------------------------------------
__global__ void ksum_kernel(const float* __restrict__ Kmat,
                            float* __restrict__ Ksum) {
    __shared__ float red[256];
    int d = blockIdx.x;
    float s = 0.0f;
    for (int t = threadIdx.x; t < TLEN; t += 256) s += Kmat[d * TLEN + t];
    red[threadIdx.x] = s;
    __syncthreads();
    for (int w = 128; w > 0; w >>= 1) {
        if (threadIdx.x < w) red[threadIdx.x] += red[threadIdx.x + w];
        __syncthreads();
    }
    if (threadIdx.x == 0) Ksum[d] = red[0];
}

// ---------------------------------------------------------------------------
// Kernel 4: R[128x512] = K[128x4096] @ X[4096x512] with V_WMMA_F32_16X16X4_F32
// X[t,(c,b)] is exactly in_chan viewed as [CB][TLEN] transposed, so row
// cb of "B" streams contiguous memory. One wave per 16x16 output tile.
// A layout (16x4 f32): lane m=lane&15 holds row d0+m; vgpr {0,1} hold
//   k = {2h, 2h+1} with h = lane>>4.
// C/D layout: vgpr v -> M = v + 8*h, N = lane&15.
// ---------------------------------------------------------------------------
__global__ void __launch_bounds__(32)
gemm_wmma_kernel(const float* __restrict__ Kmat,
                 const float* __restrict__ X,     // = in_chan [CB][TLEN]
                 float* __restrict__ R) {
    const int lane = threadIdx.x;
    const int m = lane & 15;
    const int h = lane >> 4;
    const int d0 = blockIdx.y * 16;
    const int n0 = blockIdx.x * 16;

    const float* Arow = Kmat + (size_t)(d0 + m) * TLEN + 2 * h;
    const float* Brow = X    + (size_t)(n0 + m) * TLEN + 2 * h;

    v8f acc = {};
#pragma unroll 4
    for (int k0 = 0; k0 < TLEN; k0 += 4) {
        v2f a = *(const v2f*)(Arow + k0);
        v2f b = *(const v2f*)(Brow + k0);
        // 8 args: (neg_a, A, neg_b, B, c_mod, C, reuse_a, reuse_b)
        acc = __builtin_amdgcn_wmma_f32_16x16x4_f32(
            false, a, false, b, (short)0, acc, false, false);
    }

#pragma unroll
    for (int v = 0; v < 8; ++v) {
        int dd = d0 + v + 8 * h;
        R[(size_t)dd * CB + n0 + m] = acc[v];
    }
}

// ---------------------------------------------------------------------------
// Kernel 5: y_last + GELU.
// Y[b,d] = gelu( sum_c mask[b,c]*W_in[c,d]*R[d,c*B+b]
//                + b_in[d]*Ksum[d] + D_ssm[d]*u_last[b,d] )
// u_last[b,d] = sum_c mask[b,c]^2 * W_in[c,d] + b_in[d]
// ---------------------------------------------------------------------------
__global__ void ylast_kernel(const float* __restrict__ in_chan,
                             const float* __restrict__ W_in,
                             const float* __restrict__ b_in,
                             const float* __restrict__ D_ssm,
                             const float* __restrict__ Ksum,
                             const float* __restrict__ R,
                             float* __restrict__ Y) {
    int i = blockIdx.x * blockDim.x + threadIdx.x;  // i = b*DMODEL + d
    if (i >= BATCH * DMODEL) return;
    int b = i / DMODEL;
    int d = i % DMODEL;
    float s = 0.0f, ul = 0.0f;
#pragma unroll
    for (int c = 0; c < C_IN; ++c) {
        float mk = in_chan[((size_t)c * BATCH + b) * TLEN + (TLEN - 1)];
        float w  = W_in[c * DMODEL + d];
        s  += mk * w * R[(size_t)d * CB + c * BATCH + b];
        ul += mk * mk * w;
    }
    ul += b_in[d];
    float y = s + b_in[d] * Ksum[d] + D_ssm[d] * ul;
    // jax.nn.gelu default (tanh approximation)
    float y3 = y * y * y;
    float g = 0.5f * y * (1.0f + tanhf(0.7978845608028654f * (y + 0.044715f * y3)));
    Y[i] = g;
}

// ---------------------------------------------------------------------------
// Kernel 6: head. Per b: z = Y[b]@W_mu + b_mu;  out[b] = sigmoid(z@W_lin+b_lin)
// ---------------------------------------------------------------------------
__global__ void __launch_bounds__(HID)
head_kernel(const float* __restrict__ Y,
            const float* __restrict__ W_mu,
            const float* __restrict__ b_mu,
            const float* __restrict__ W_lin,
            const float* __restrict__ b_lin,
            float* __restrict__ out) {
    __shared__ float yg[DMODEL];
    __shared__ float red[HID];
    int b  = blockIdx.x;
    int hh = threadIdx.x;
    yg[hh] = Y[b * DMODEL + hh];   // DMODEL == HID == 128
    __syncthreads();
    float z = b_mu[hh];
#pragma unroll 8
    for (int d = 0; d < DMODEL; ++d) z += yg[d] * W_mu[d * HID + hh];
    red[hh] = z * W_lin[hh];
    __syncthreads();
    for (int w = HID / 2; w > 0; w >>= 1) {
        if (hh < w) red[hh] += red[hh + w];
        __syncthreads();
    }
    if (hh == 0) out[b] = 1.0f / (1.0f + __expf(-(red[0] + b_lin[0])));
}

// ---------------------------------------------------------------------------
// Launcher. Input order per setup_inputs():
// 0 in_chan, 1 h_0, 2 c_0, 3 W_in, 4 b_in, 5 log_a, 6 B_ssm, 7 C_ssm,
// 8 D_ssm, 9 W_mu, 10 b_mu, 11 W_lin, 12 b_lin.  Output: 64 floats.
// ---------------------------------------------------------------------------
extern "C" void kernel_launch(void* const* d_in, const int* in_sizes, int n_in,
                              void* d_out, int out_size, void* d_ws, size_t ws_size,
                              hipStream_t stream) {
    const float* in_chan = (const float*)d_in[0];
    const float* W_in  = (const float*)d_in[3];
    const float* b_in  = (const float*)d_in[4];
    const float* log_a = (const float*)d_in[5];
    const float* B_ssm = (const float*)d_in[6];
    const float* C_ssm = (const float*)d_in[7];
    const float* D_ssm = (const float*)d_in[8];
    const float* W_mu  = (const float*)d_in[9];
    const float* b_mu  = (const float*)d_in[10];
    const float* W_lin = (const float*)d_in[11];
    const float* b_lin = (const float*)d_in[12];
    float* out = (float*)d_out;

    // Workspace layout (floats). Everything is written before it is read;
    // no dependence on prior contents (harness poisons ws with 0xAA).
    float* ws   = (float*)d_ws;
    float* G    = ws;                              //  8192
    float* L2A  = ws + 8192;                       //  8192
    float* Ksum = ws + 16384;                      //   128
    float* Kmat = ws + 16512;                      // 524288 (16B-aligned)
    float* R    = Kmat + (size_t)DMODEL * TLEN;    //  65536
    float* Y    = R + (size_t)DMODEL * CB;         //   8192
    // total ~2.46 MB

    prep_kernel<<<(DMODEL * NSTATE) / 256, 256, 0, stream>>>(log_a, B_ssm, C_ssm, G, L2A);
    kfilter_kernel<<<(DMODEL * TLEN) / 256, 256, 0, stream>>>(G, L2A, Kmat);
    ksum_kernel<<<DMODEL, 256, 0, stream>>>(Kmat, Ksum);

    dim3 ggrid(CB / 16, DMODEL / 16);              // 32 x 8 wave-tiles
    gemm_wmma_kernel<<<ggrid, 32, 0, stream>>>(Kmat, in_chan, R);

    ylast_kernel<<<(BATCH * DMODEL) / 256, 256, 0, stream>>>(
        in_chan, W_in, b_in, D_ssm, Ksum, R, Y);
    head_kernel<<<BATCH, HID, 0, stream>>>(Y, W_mu, b_mu, W_lin, b_lin, out);
}